// multi_scale_smoothing_10453950398775
// MI455X (gfx1250) — compile-verified
//
#include <hip/hip_runtime.h>
#include <hip/hip_bf16.h>
#include <stdint.h>

// Problem constants (from the reference): x is (64, 65536) f32, SCALE=8.
#define ROWS   64
#define LEN    65536
#define FRONT  7            // SCALE-1 halo on each side
#define TPB    256          // 8 wave32s per block
#define CHUNK  8            // elements per thread
#define TILE   (TPB * CHUNK)            // 2048 elements per block
#define LDS_N  (9 * TPB)                // 2304 >= TILE + 2*FRONT, uniform 9 loads/thread

// Native clang vector type so __builtin_nontemporal_store emits b128 stores.
typedef __attribute__((ext_vector_type(4))) float v4f;

__global__ __launch_bounds__(TPB)
void multi_scale_smoothing_kernel(const float* __restrict__ x,
                                  float* __restrict__ out) {
    __shared__ float lds[LDS_N];

    const int tid    = threadIdx.x;
    const int tileIx = blockIdx.x;          // 0..31
    const int row    = blockIdx.y;          // 0..63
    const int base_l = tileIx * TILE;       // first element index of this tile

    const float* rowp = x + (size_t)row * LEN;

    // 32-bit wave-relative LDS byte offset of lds[0] (low 32 bits of the
    // generic address of an LDS object are the LDS offset on gfx1250).
    const uint32_t lds0 = (uint32_t)(uintptr_t)(&lds[0]);

    // ---- Stage tile + halo into LDS via CDNA5 async global->LDS copies ----
    // lds[i] = row[clamp(base_l - FRONT + i)]  (edge padding == clamp)
    #pragma unroll
    for (int k = 0; k < LDS_N / TPB; ++k) {
        const int i = k * TPB + tid;
        int l = base_l - FRONT + i;
        l = l < 0 ? 0 : (l > LEN - 1 ? LEN - 1 : l);
        const uint32_t goff = (uint32_t)l * 4u;        // byte offset into row
        const uint32_t loff = lds0 + (uint32_t)i * 4u; // LDS byte address
        // GLOBAL_LOAD_ASYNC_TO_LDS_B32, GVS mode: copies the dword at saddr + vaddr to LDS address vdst
        asm volatile("global_load_async_to_lds_b32 %0, %1, %2"
                     :: "v"(loff), "v"(goff), "s"(rowp)
                     : "memory");
    }
    // Wait for this wave's async copies, then workgroup barrier so every
    // wave sees the whole tile.
    asm volatile("s_wait_asynccnt 0" ::: "memory");
    __syncthreads();

    // ---- Compute: running box sums, widths 1..15, from LDS ----
    // Reciprocal-multiply instead of f32 divide (divide chains would cost
    // more time than the 144MB of memory traffic on this chip).
    #pragma unroll
    for (int k = 0; k < CHUNK; ++k) {
        const int i  = k * TPB + tid;     // element index within tile
        const int c  = i + FRONT;         // center index in LDS

        float s = lds[c];
        v4f o0, o1;
        o0.x = s;
        s += lds[c - 1] + lds[c + 1];  o0.y = s * (1.0f / 3.0f);
        s += lds[c - 2] + lds[c + 2];  o0.z = s * (1.0f / 5.0f);
        s += lds[c - 3] + lds[c + 3];  o0.w = s * (1.0f / 7.0f);
        s += lds[c - 4] + lds[c + 4];  o1.x = s * (1.0f / 9.0f);
        s += lds[c - 5] + lds[c + 5];  o1.y = s * (1.0f / 11.0f);
        s += lds[c - 6] + lds[c + 6];  o1.z = s * (1.0f / 13.0f);
        s += lds[c - 7] + lds[c + 7];  o1.w = s * (1.0f / 15.0f);

        // out[row, l, 0..7] -> 32 contiguous bytes; consecutive lanes write
        // consecutive 32B chunks (fully coalesced b128 stores).
        const size_t l = (size_t)base_l + (size_t)i;
        v4f* op = (v4f*)(out + ((size_t)row * LEN + l) * 8u);
        __builtin_nontemporal_store(o0, op + 0);
        __builtin_nontemporal_store(o1, op + 1);
    }
}

extern "C" void kernel_launch(void* const* d_in, const int* in_sizes, int n_in,
                              void* d_out, int out_size, void* d_ws, size_t ws_size,
                              hipStream_t stream) {
    (void)in_sizes; (void)n_in; (void)out_size; (void)d_ws; (void)ws_size;
    const float* x   = (const float*)d_in[0];
    float*       out = (float*)d_out;
    dim3 grid(LEN / TILE, ROWS, 1);   // (32, 64)
    dim3 block(TPB, 1, 1);
    multi_scale_smoothing_kernel<<<grid, block, 0, stream>>>(x, out);
}